// computeInsulation_14328010900237
// MI455X (gfx1250) — compile-verified
//
#include <hip/hip_runtime.h>
#include <hip/hip_bf16.h>

typedef __attribute__((ext_vector_type(2))) float v2f;
typedef __attribute__((ext_vector_type(8))) float v8f;

#define NSIDE   2048
#define KWIN    17          // 2*WINDOW_RADIUS + 1
#define KELEMS  289         // 17*17
#define LIV     2032        // NSIDE - KWIN + 1
#define LDV     2017        // LIV - 8 - 8 + 1  (deriv windows)
#define LBAND   2015
#define BATCH   16
#define TOT_IV  (BATCH * LIV)        // 32512
#define WAVES_PER_BATCH (LIV / 16)   // 127
#define TOT_WAVES (BATCH * WAVES_PER_BATCH) // 2032

// ---------------------------------------------------------------------------
// Kernel A: per-wave, 16 diagonal 17x17 block sums via V_WMMA_F32_16X16X4_F32.
// A rows = 16 outputs, K = 4 flattened block elements per step, B = ones, so
// D = A*1 + C accumulates the row sums; 73 steps cover 289 elements (tail=0).
// A layout (ISA 7.12.2, 32-bit A 16x4): lanes 0-15 -> M, VGPR0=K0/VGPR1=K1;
// lanes 16-31 -> M, VGPR0=K2/VGPR1=K3.  D layout: lane 0 = M0..7, lane16 = M8..15.
// ---------------------------------------------------------------------------
__global__ void block_sums_kernel(const float* __restrict__ x,
                                  float* __restrict__ sums) {
  const int lane = threadIdx.x & 31;
  const int wave = (blockIdx.x * blockDim.x + threadIdx.x) >> 5;  // 0..2031
  const int b    = wave / WAVES_PER_BATCH;
  const int i0   = (wave % WAVES_PER_BATCH) * 16;
  const float* xb = x + (size_t)b * NSIDE * NSIDE;

  const int m    = lane & 15;      // M row == which of my wave's 16 outputs
  const int half = lane >> 4;      // 0: K=0,1  1: K=2,3
  const int i    = i0 + m;                              // output index
  const size_t base = (size_t)i * NSIDE + i;            // block top-left

  __builtin_prefetch(&xb[base], 0, 1);   // global_prefetch_b8 (band is L2-hot)

#if __has_builtin(__builtin_amdgcn_wmma_f32_16x16x4_f32)
  v8f c = {};
  const v2f bones = {1.0f, 1.0f};        // layout-agnostic all-ones B matrix
  for (int t = 0; t < 73; ++t) {
    const int e0 = 4 * t + 2 * half;     // flattened element -> K = 2*half(+1)
    const int e1 = e0 + 1;
    float a0 = 0.0f, a1 = 0.0f;
    if (e0 < KELEMS) a0 = xb[base + (size_t)(e0 / KWIN) * NSIDE + (e0 % KWIN)];
    if (e1 < KELEMS) a1 = xb[base + (size_t)(e1 / KWIN) * NSIDE + (e1 % KWIN)];
    v2f a = {a0, a1};
    c = __builtin_amdgcn_wmma_f32_16x16x4_f32(
        /*neg_a=*/false, a, /*neg_b=*/false, bones,
        /*c_mod=*/(short)0, c, /*reuse_a=*/false, /*reuse_b=*/false);
  }
  // All D columns are identical; lane 0 holds M=0..7, lane 16 holds M=8..15.
  if (m == 0) {
    float* dst = sums + b * LIV + i0 + half * 8;
#pragma unroll
    for (int v = 0; v < 8; ++v) dst[v] = c[v];
  }
#else
  if (lane < 16) {
    float s = 0.0f;
    for (int e = 0; e < KELEMS; ++e)
      s += xb[base + (size_t)(e / KWIN) * NSIDE + (e % KWIN)];
    sums[b * LIV + i] = s;
  }
#endif
}

// ---------------------------------------------------------------------------
// Kernel B: deterministic single-block reduction of all 32512 raw block sums.
// iv = log2((sum/289)/mean) = log2(sum * TOT_IV / total)  (289 cancels),
// so store scale = TOT_IV / total.
// ---------------------------------------------------------------------------
__global__ void reduce_scale_kernel(const float* __restrict__ sums,
                                    float* __restrict__ scale) {
  __shared__ float sdata[256];
  float s = 0.0f;
  for (int idx = threadIdx.x; idx < TOT_IV; idx += 256) s += sums[idx];
  sdata[threadIdx.x] = s;
  __syncthreads();
  for (int off = 128; off > 0; off >>= 1) {
    if (threadIdx.x < off) sdata[threadIdx.x] += sdata[threadIdx.x + off];
    __syncthreads();
  }
  if (threadIdx.x == 0) scale[0] = (float)TOT_IV / sdata[0];
}

// ---------------------------------------------------------------------------
// Kernel C: iv = log2(sum * scale)
// ---------------------------------------------------------------------------
__global__ void iv_kernel(const float* __restrict__ sums,
                          const float* __restrict__ scale,
                          float* __restrict__ out_iv) {
  const int idx = blockIdx.x * blockDim.x + threadIdx.x;
  if (idx < TOT_IV) out_iv[idx] = __log2f(sums[idx] * scale[0]);
}

// ---------------------------------------------------------------------------
// Kernel D: dv[i] = (sum iv[i+8..i+15] - sum iv[i..i+7]) / 8
//           band[i] = (dv[i] < 0) & (dv[i+2] > 0)   (dv[i+2] recomputed)
// ---------------------------------------------------------------------------
__global__ void dv_band_kernel(const float* __restrict__ iv,
                               float* __restrict__ out_dv,
                               float* __restrict__ out_band) {
  const int idx = blockIdx.x * blockDim.x + threadIdx.x;
  if (idx >= BATCH * LDV) return;
  const int b = idx / LDV;
  const int i = idx % LDV;
  const float* ivb = iv + b * LIV;

  float bot = 0.0f, top = 0.0f;
#pragma unroll
  for (int j = 0; j < 8; ++j) { bot += ivb[i + j]; top += ivb[i + 8 + j]; }
  const float dvA = (top - bot) * 0.125f;
  out_dv[b * LDV + i] = dvA;

  if (i < LBAND) {
    float bot2 = 0.0f, top2 = 0.0f;
#pragma unroll
    for (int j = 0; j < 8; ++j) { bot2 += ivb[i + 2 + j]; top2 += ivb[i + 10 + j]; }
    const float dvB = (top2 - bot2) * 0.125f;
    out_band[b * LBAND + i] = (dvA < 0.0f && dvB > 0.0f) ? 1.0f : 0.0f;
  }
}

// ---------------------------------------------------------------------------
extern "C" void kernel_launch(void* const* d_in, const int* in_sizes, int n_in,
                              void* d_out, int out_size, void* d_ws, size_t ws_size,
                              hipStream_t stream) {
  const float* x = (const float*)d_in[0];
  float* out = (float*)d_out;

  float* sums  = (float*)d_ws;        // TOT_IV raw block sums
  float* scale = sums + TOT_IV;       // 1 float

  float* out_iv   = out;                          // 16*2032
  float* out_dv   = out + TOT_IV;                 // 16*2017
  float* out_band = out + TOT_IV + BATCH * LDV;   // 16*2015

  // 2032 waves, 8 waves (256 threads) per block -> exactly 254 blocks.
  block_sums_kernel<<<TOT_WAVES / 8, 256, 0, stream>>>(x, sums);
  reduce_scale_kernel<<<1, 256, 0, stream>>>(sums, scale);
  iv_kernel<<<(TOT_IV + 255) / 256, 256, 0, stream>>>(sums, scale, out_iv);
  dv_band_kernel<<<(BATCH * LDV + 255) / 256, 256, 0, stream>>>(out_iv, out_dv, out_band);
}